// TransformLayer_29789893165586
// MI455X (gfx1250) — compile-verified
//
#include <hip/hip_runtime.h>
#include <cstddef>

typedef float v2f __attribute__((ext_vector_type(2)));
typedef float v8f __attribute__((ext_vector_type(8)));

#define NJ 20

__device__ __forceinline__ void rpy2r_dev(float r, float p, float y, float* R) {
  float cr = cosf(r), sr = sinf(r);
  float cp = cosf(p), sp = sinf(p);
  float cy = cosf(y), sy = sinf(y);
  R[0] = cy * cp; R[1] = cy * sp * sr - sy * cr; R[2] = cy * sp * cr + sy * sr;
  R[3] = sy * cp; R[4] = sy * sp * sr + cy * cr; R[5] = sy * sp * cr - cy * sr;
  R[6] = -sp;     R[7] = cp * sr;                R[8] = cp * cr;
}

__device__ __forceinline__ void mat3mul(const float* A, const float* B, float* C) {
#pragma unroll
  for (int i = 0; i < 3; ++i)
#pragma unroll
    for (int k = 0; k < 3; ++k)
      C[i * 3 + k] = A[i * 3 + 0] * B[0 * 3 + k] + A[i * 3 + 1] * B[1 * 3 + k] +
                     A[i * 3 + 2] * B[2 * 3 + k];
}

// cst layout per joint (stride 64 floats):
// [0:9) Ror   [9:12) por   [12:21) Rop   [21:24) pop
// [24:27) rev_axis (normalized)   [27:30) pri_axis
// [30:46) T_track 4x4 row-major
__global__ void fk_const_kernel(const float* __restrict__ rev_p_off,
                                const float* __restrict__ rev_rpy_off,
                                const float* __restrict__ pri_p_off,
                                const float* __restrict__ pri_rpy_off,
                                const float* __restrict__ rev_axis,
                                const float* __restrict__ pri_axis,
                                const float* __restrict__ p_track,
                                const float* __restrict__ rpy_track,
                                float* __restrict__ cst) {
  int j = threadIdx.x;
  if (j >= NJ) return;
  float* o = cst + j * 64;
  float Rt[9];

  rpy2r_dev(rev_rpy_off[3 * j], rev_rpy_off[3 * j + 1], rev_rpy_off[3 * j + 2], Rt);
#pragma unroll
  for (int k = 0; k < 9; ++k) o[k] = Rt[k];
  o[9] = rev_p_off[3 * j]; o[10] = rev_p_off[3 * j + 1]; o[11] = rev_p_off[3 * j + 2];

  rpy2r_dev(pri_rpy_off[3 * j], pri_rpy_off[3 * j + 1], pri_rpy_off[3 * j + 2], Rt);
#pragma unroll
  for (int k = 0; k < 9; ++k) o[12 + k] = Rt[k];
  o[21] = pri_p_off[3 * j]; o[22] = pri_p_off[3 * j + 1]; o[23] = pri_p_off[3 * j + 2];

  float x = rev_axis[3 * j], y = rev_axis[3 * j + 1], z = rev_axis[3 * j + 2];
  float inv = 1.f / (sqrtf(x * x + y * y + z * z) + 1e-12f);
  o[24] = x * inv; o[25] = y * inv; o[26] = z * inv;

  o[27] = pri_axis[3 * j]; o[28] = pri_axis[3 * j + 1]; o[29] = pri_axis[3 * j + 2];

  rpy2r_dev(rpy_track[3 * j], rpy_track[3 * j + 1], rpy_track[3 * j + 2], Rt);
  o[30] = Rt[0]; o[31] = Rt[1]; o[32] = Rt[2]; o[33] = p_track[3 * j];
  o[34] = Rt[3]; o[35] = Rt[4]; o[36] = Rt[5]; o[37] = p_track[3 * j + 1];
  o[38] = Rt[6]; o[39] = Rt[7]; o[40] = Rt[8]; o[41] = p_track[3 * j + 2];
  o[42] = 0.f; o[43] = 0.f; o[44] = 0.f; o[45] = 1.f;
}

__global__ __launch_bounds__(256) void fk_main_kernel(
    const float* __restrict__ rev_q, const float* __restrict__ pri_q,
    const float* __restrict__ cst, float* __restrict__ out, int B) {
  __shared__ float stage[8 * 512];  // 512 floats (32 lanes x 4x4) per wave

  const int tid  = blockIdx.x * 256 + threadIdx.x;  // one batch chain per lane
  const int lane = threadIdx.x & 31;
  const int wv   = threadIdx.x >> 5;
  float* st = &stage[wv * 512];

  const bool alive = (tid < B);
  const int  bs    = alive ? tid : (B - 1);

  float* trk_out = out;
  float* rev_out = out + (size_t)B * 160;
  float* pri_out = out + (size_t)B * 480;

  // WMMA lane geometry (wave32): A/B halves, D extraction
  const int  h       = lane >> 4;         // lane half
  const int  m16     = lane & 15;
  const int  nu      = m16 & 3;           // matrix column / A row-within-batch
  const int  iq      = m16 >> 2;          // batch quarter within group of 4
  const bool useful  = (h == (iq >> 1));  // this lane owns unique D elements
  const int  baseSel = (iq & 1) << 2;     // D vgpr base for this batch's rows
  const long waveBase = (long)blockIdx.x * 256 + (long)wv * 32;

  float R[9] = {1.f, 0.f, 0.f, 0.f, 1.f, 0.f, 0.f, 0.f, 1.f};
  float p[3] = {0.f, 0.f, 0.f};

  for (int j = 0; j < NJ; ++j) {
    const float* C = cst + j * 64;
    float Ror[9], por[3], Rop[9], pop[3], ar[3], ap[3];
#pragma unroll
    for (int k = 0; k < 9; ++k) Ror[k] = C[k];
#pragma unroll
    for (int k = 0; k < 3; ++k) por[k] = C[9 + k];
#pragma unroll
    for (int k = 0; k < 9; ++k) Rop[k] = C[12 + k];
#pragma unroll
    for (int k = 0; k < 3; ++k) pop[k] = C[21 + k];
#pragma unroll
    for (int k = 0; k < 3; ++k) ar[k] = C[24 + k];
#pragma unroll
    for (int k = 0; k < 3; ++k) ap[k] = C[27 + k];

    float q = rev_q[(size_t)bs * NJ + j];
    float d = pri_q[(size_t)bs * NJ + j];

    // Rodrigues: R = I + sin(q) K + (1-cos(q)) K^2, axis pre-normalized
    float sn = sinf(q), cs = cosf(q), oc = 1.f - cs;
    float ax = ar[0], ay = ar[1], az = ar[2];
    float Rod[9];
    Rod[0] = 1.f + oc * (-(ay * ay + az * az));
    Rod[1] = -sn * az + oc * (ax * ay);
    Rod[2] =  sn * ay + oc * (ax * az);
    Rod[3] =  sn * az + oc * (ax * ay);
    Rod[4] = 1.f + oc * (-(ax * ax + az * az));
    Rod[5] = -sn * ax + oc * (ay * az);
    Rod[6] = -sn * ay + oc * (ax * az);
    Rod[7] =  sn * ax + oc * (ay * az);
    Rod[8] = 1.f + oc * (-(ax * ax + ay * ay));

    // out1 = out @ (rev_T_off @ rev_T) : rev_step = (Ror@Rod, por)
    float Rrs[9]; mat3mul(Ror, Rod, Rrs);
    float R1[9];  mat3mul(R, Rrs, R1);
    float p1[3];
    p1[0] = R[0] * por[0] + R[1] * por[1] + R[2] * por[2] + p[0];
    p1[1] = R[3] * por[0] + R[4] * por[1] + R[5] * por[2] + p[1];
    p1[2] = R[6] * por[0] + R[7] * por[1] + R[8] * por[2] + p[2];

    if (alive) {
      float4* o4 = (float4*)(rev_out + ((size_t)tid * NJ + j) * 16);
      o4[0] = make_float4(R1[0], R1[1], R1[2], p1[0]);
      o4[1] = make_float4(R1[3], R1[4], R1[5], p1[1]);
      o4[2] = make_float4(R1[6], R1[7], R1[8], p1[2]);
      o4[3] = make_float4(0.f, 0.f, 0.f, 1.f);
    }

    // out2 = out1 @ (pri_T_off @ pri_T) : pri_step = (Rop, Rop@(d*ap)+pop)
    float tv0 = d * ap[0], tv1 = d * ap[1], tv2 = d * ap[2];
    float ppv[3];
    ppv[0] = Rop[0] * tv0 + Rop[1] * tv1 + Rop[2] * tv2 + pop[0];
    ppv[1] = Rop[3] * tv0 + Rop[4] * tv1 + Rop[5] * tv2 + pop[1];
    ppv[2] = Rop[6] * tv0 + Rop[7] * tv1 + Rop[8] * tv2 + pop[2];
    float R2[9]; mat3mul(R1, Rop, R2);
    float p2[3];
    p2[0] = R1[0] * ppv[0] + R1[1] * ppv[1] + R1[2] * ppv[2] + p1[0];
    p2[1] = R1[3] * ppv[0] + R1[4] * ppv[1] + R1[5] * ppv[2] + p1[1];
    p2[2] = R1[6] * ppv[0] + R1[7] * ppv[1] + R1[8] * ppv[2] + p1[2];
#pragma unroll
    for (int k = 0; k < 9; ++k) R[k] = R2[k];
    p[0] = p2[0]; p[1] = p2[1]; p[2] = p2[2];

    if (alive) {
      float4* o4 = (float4*)(pri_out + ((size_t)tid * NJ + j) * 16);
      o4[0] = make_float4(R[0], R[1], R[2], p[0]);
      o4[1] = make_float4(R[3], R[4], R[5], p[1]);
      o4[2] = make_float4(R[6], R[7], R[8], p[2]);
      o4[3] = make_float4(0.f, 0.f, 0.f, 1.f);
    }

    // Branch joints (even j): TrackingSE3 = out2 @ T_track via WMMA f32 16x16x4.
    // A (16x4) = 4 batches' out2 stacked; B (4x16) = T_track replicated 4x.
    if ((j & 1) == 0) {
      __syncthreads();  // protect staging buffer vs previous iteration's loads
      float* sl = st + lane * 16;  // this lane's out2, row-major 4x4
      sl[0] = R[0]; sl[1] = R[1]; sl[2]  = R[2]; sl[3]  = p[0];
      sl[4] = R[3]; sl[5] = R[4]; sl[6]  = R[5]; sl[7]  = p[1];
      sl[8] = R[6]; sl[9] = R[7]; sl[10] = R[8]; sl[11] = p[2];
      sl[12] = 0.f; sl[13] = 0.f; sl[14] = 0.f; sl[15] = 1.f;
      __syncthreads();

      // B operand: vgpr v, half h -> K = v + 2h, column = nu
      v2f Bop;
      Bop.x = C[30 + (2 * h + 0) * 4 + nu];
      Bop.y = C[30 + (2 * h + 1) * 4 + nu];

      const v8f zero8 = {0.f, 0.f, 0.f, 0.f, 0.f, 0.f, 0.f, 0.f};
#pragma unroll
      for (int g = 0; g < 8; ++g) {
        // A operand: lane holds row m16 of the stack -> batch (4g+iq), row nu, K = 2h,2h+1
        const v2f A = *(const v2f*)(st + (g * 4 + iq) * 16 + nu * 4 + 2 * h);
        v8f D = __builtin_amdgcn_wmma_f32_16x16x4_f32(
            false, A, false, Bop, (short)0, zero8, false, false);
        // D vgpr r, half h -> element (M = r + 8h, N = m16); lane owns batch iq's
        // rows 4*iq..4*iq+3 (at vgprs baseSel..baseSel+3) iff half matches.
        float e0 = baseSel ? D[4] : D[0];
        float e1 = baseSel ? D[5] : D[1];
        float e2 = baseSel ? D[6] : D[2];
        float e3 = baseSel ? D[7] : D[3];
        long bb = waveBase + g * 4 + iq;
        if (useful && bb < (long)B) {
          float* o = trk_out + ((size_t)bb * 10 + (j >> 1)) * 16 + nu;
          o[0] = e0; o[4] = e1; o[8] = e2; o[12] = e3;
        }
      }
    }
  }
}

extern "C" void kernel_launch(void* const* d_in, const int* in_sizes, int n_in,
                              void* d_out, int out_size, void* d_ws, size_t ws_size,
                              hipStream_t stream) {
  (void)n_in; (void)out_size; (void)ws_size;
  const float* rev_q       = (const float*)d_in[0];
  const float* pri_q       = (const float*)d_in[1];
  const float* rev_p_off   = (const float*)d_in[2];
  const float* rev_rpy_off = (const float*)d_in[3];
  const float* pri_p_off   = (const float*)d_in[4];
  const float* pri_rpy_off = (const float*)d_in[5];
  const float* rev_axis    = (const float*)d_in[6];
  const float* pri_axis    = (const float*)d_in[7];
  const float* p_track     = (const float*)d_in[8];
  const float* rpy_track   = (const float*)d_in[9];

  float* cst = (float*)d_ws;  // 20 * 64 floats = 5 KB
  int B = in_sizes[0] / NJ;

  fk_const_kernel<<<1, 32, 0, stream>>>(rev_p_off, rev_rpy_off, pri_p_off, pri_rpy_off,
                                        rev_axis, pri_axis, p_track, rpy_track, cst);
  int grid = (B + 255) / 256;
  fk_main_kernel<<<grid, 256, 0, stream>>>(rev_q, pri_q, cst, (float*)d_out, B);
}